// GaborConv1D_51531017617939
// MI455X (gfx1250) — compile-verified
//
#include <hip/hip_runtime.h>
#include <hip/hip_bf16.h>

typedef __attribute__((ext_vector_type(2))) float v2f;
typedef __attribute__((ext_vector_type(8))) float v8f;

#define T_LEN   16384
#define TP_LEN  16448    /* T + 2*PAD */
#define HOP     32
#define NFRM    513
#define NFRMP   514      /* padded frame count (tile-exact)                 */
#define NB      8
#define NCH     64
#define NFREQ   33
#define COL2P   4112     /* NB*NFRMP cols per ri half for mixing GEMM       */
#define COL1P   263168   /* 512*NFRMP cols for STFT/ISTFT GEMMs             */
#define PLANE   263168   /* per-fr plane: U2 [64ic x 4112] = Y2 [512bo x 514] */

/* ws layout (floats) */
#define OFF_A1  0
#define OFF_A3  4096
#define OFF_WT  8192                        /* 33*4096 = 135168 */
#define OFF_U   143360                      /* 64*PLANE = 16842752 */
#define OFF_Y   (OFF_U + 64*PLANE)
#define OFF_FR  OFF_U                       /* frames alias U after mixing   */
#define OFF_XP  OFF_Y                       /* padded x aliases Y before mix */

__device__ __forceinline__ v8f wmma_f32(v2f a, v2f b, v8f c) {
    return __builtin_amdgcn_wmma_f32_16x16x4_f32(false, a, false, b, (short)0, c,
                                                 false, false);
}

/* ---- build windowed DFT / iDFT matrices + transposed weight ------------- */
__global__ void prep_kernel(const float* __restrict__ wgt,
                            float* __restrict__ A1, float* __restrict__ A3,
                            float* __restrict__ Wt) {
    const int gid = blockIdx.x * blockDim.x + threadIdx.x;
    const int gsz = gridDim.x * blockDim.x;
    const float PI = 3.14159265358979f;
    for (int idx = gid; idx < 4096; idx += gsz) {          /* A1: rows fr, cols tau */
        int fr = idx >> 6, tau = idx & 63;
        float wv = 0.5f * (1.0f - cosf(2.0f * PI * (float)tau / 64.0f));
        float v;
        if (fr <= 32) v =  cosf(2.0f * PI * (float)(fr)      * (float)tau / 64.0f) * wv;
        else          v = -sinf(2.0f * PI * (float)(fr - 32) * (float)tau / 64.0f) * wv;
        A1[idx] = v;
    }
    for (int idx = gid; idx < 4096; idx += gsz) {          /* A3: rows tau, cols fr */
        int tau = idx >> 6, fr = idx & 63;
        float wv = 0.5f * (1.0f - cosf(2.0f * PI * (float)tau / 64.0f));
        float s = wv * (1.0f / 64.0f);
        float v;
        if (fr == 0)       v = s;
        else if (fr < 32)  v = 2.0f * cosf(2.0f * PI * (float)fr * (float)tau / 64.0f) * s;
        else if (fr == 32) v = cosf(PI * (float)tau) * s;
        else               v = -2.0f * sinf(2.0f * PI * (float)(fr - 32) * (float)tau / 64.0f) * s;
        A3[idx] = v;
    }
    for (int idx = gid; idx < NFREQ * 4096; idx += gsz) {  /* Wt[f][o][i] */
        int f = idx >> 12, oi = idx & 4095;
        Wt[idx] = wgt[(size_t)oi * NFREQ + f];
    }
}

/* ---- reflect-pad x once: xp[bc][tp], tp in [0,16448) -------------------- */
__global__ void pad_kernel(const float* __restrict__ x, float* __restrict__ xp) {
    const int tp = blockIdx.x * blockDim.x + threadIdx.x;
    if (tp >= TP_LEN) return;
    const int bc = blockIdx.y;
    int t = tp - 32;
    t = t < 0 ? -t : t;
    t = t >= T_LEN ? 2 * T_LEN - 2 - t : t;
    xp[(size_t)bc * TP_LEN + tp] = x[(size_t)bc * T_LEN + t];
}

/* ---- GEMM1: packed rfft of windowed frames -> U2[fr][ic][b*514+n] ------- */
__global__ void stft_gemm(const float* __restrict__ xp, const float* __restrict__ A1,
                          float* __restrict__ U) {
    __shared__ float lA[4096];
    const int tid = threadIdx.x;
#pragma unroll
    for (int i = 0; i < 8; ++i)                /* 16KB A1 -> LDS, coalesced b128 */
        *(float4*)(lA + (i * 128 + tid) * 4) =
            *(const float4*)(A1 + (i * 128 + tid) * 4);
    __syncthreads();

    const int lane = tid & 31;
    const int w    = tid >> 5;                 /* M tile 0..3 */
    const int cL   = lane & 15;
    const int hs   = lane >> 4;
    const int colp = blockIdx.x * 16 + cL;     /* (b*64+c)*514 + n, n<514 */
    const int n    = colp % NFRMP;
    const int bc   = colp / NFRMP;
    const int b    = bc >> 6;
    const int ic   = bc & 63;
    const int nc   = n > 512 ? 512 : n;        /* clamp padding column */
    const float* xrow = xp + (size_t)bc * TP_LEN + nc * HOP;
    const int rowA = 16 * w + cL;
    v8f c = {};
#pragma unroll
    for (int k0 = 0; k0 < 64; k0 += 4) {
        const int kk = k0 + 2 * hs;
        v2f a  = *(const v2f*)(lA + rowA * 64 + kk);   /* ds_load_b64, imm offs */
        v2f bv = *(const v2f*)(xrow + kk);             /* global_load_b64, imm  */
        c = wmma_f32(a, bv, c);
    }
    /* packed-fr store: one lane base + r*PLANE immediate offsets */
    float* Ust = U + (size_t)(16 * w + 8 * hs) * PLANE +
                 (size_t)ic * COL2P + b * NFRMP + n;
#pragma unroll
    for (int r = 0; r < 8; ++r)
        Ust[(size_t)r * PLANE] = c[r];
}

/* ---- GEMM2: per-frequency channel mixing, ri = blockIdx.z --------------- */
__global__ void mix_gemm(const float* __restrict__ Wt, const float* __restrict__ U,
                         float* __restrict__ Y) {
    const int f  = blockIdx.y;
    const int ri = blockIdx.z;
    if (ri == 1 && (f == 0 || f == 32)) return;   /* imag f=0/32 do not exist */
    const int frq = f + ri * 32;                  /* packed-fr row, scalar    */

    __shared__ float lA[4096];
    __shared__ float lB[1024];
    const int tid = threadIdx.x;
    const float* Wf = Wt + (size_t)f * 4096;
    const float* Ub = U + (size_t)frq * PLANE + blockIdx.x * 16;
#pragma unroll
    for (int i = 0; i < 8; ++i)                /* 16KB W[f] -> LDS */
        *(float4*)(lA + (i * 128 + tid) * 4) =
            *(const float4*)(Wf + (i * 128 + tid) * 4);
#pragma unroll
    for (int i = 0; i < 2; ++i) {              /* 64x16 U tile -> LDS */
        const int fi = i * 128 + tid;          /* float4 index 0..255 */
        const int r = fi >> 2, p = fi & 3;
        *(float4*)(lB + fi * 4) = *(const float4*)(Ub + (size_t)r * COL2P + p * 4);
    }
    __syncthreads();

    const int lane = tid & 31;
    const int w    = tid >> 5;
    const int cL   = lane & 15;
    const int hs   = lane >> 4;
    const int rowA = 16 * w + cL;
    v8f c = {};
#pragma unroll
    for (int k0 = 0; k0 < 64; k0 += 4) {
        const int kk = k0 + 2 * hs;
        v2f a = *(const v2f*)(lA + rowA * 64 + kk);
        v2f bv;
        bv.x = lB[kk * 16 + cL];
        bv.y = lB[(kk + 1) * 16 + cL];
        c = wmma_f32(a, bv, c);
    }
    /* store into Y2[frq][(b*64+o)*514+n]: lane base + r*514 immediates */
    const int col2p = blockIdx.x * 16 + cL;    /* b*514 + n */
    const int bI = col2p / NFRMP;
    const int nI = col2p % NFRMP;
    float* Yst = Y + (size_t)frq * PLANE + (size_t)bI * (64 * NFRMP) + nI +
                 (size_t)(16 * w + 8 * hs) * NFRMP;
#pragma unroll
    for (int r = 0; r < 8; ++r)
        Yst[(size_t)r * NFRMP] = c[r];
}

/* ---- GEMM3: packed irfft * window -> frames[colp][tau] ------------------ */
__global__ void istft_gemm(const float* __restrict__ A3, const float* __restrict__ Y,
                           float* __restrict__ frames) {
    __shared__ float lA[4096];
    const int tid = threadIdx.x;
#pragma unroll
    for (int i = 0; i < 8; ++i)                /* 16KB A3 -> LDS */
        *(float4*)(lA + (i * 128 + tid) * 4) =
            *(const float4*)(A3 + (i * 128 + tid) * 4);
    __syncthreads();

    const int lane = tid & 31;
    const int w    = tid >> 5;
    const int cL   = lane & 15;
    const int hs   = lane >> 4;
    const int colp = blockIdx.x * 16 + cL;     /* (b*64+o)*514 + n */
    const float* Yb = Y + (size_t)colp + (size_t)(2 * hs) * PLANE;
    const int rowA = 16 * w + cL;              /* tau */
    v8f c = {};
#pragma unroll
    for (int k0 = 0; k0 < 64; k0 += 4) {
        v2f a = *(const v2f*)(lA + rowA * 64 + (k0 + 2 * hs));
        v2f bv;
        bv.x = Yb[(size_t)k0 * PLANE];         /* strength-reduced base     */
        bv.y = Yb[(size_t)k0 * PLANE + PLANE]; /* +PLANE fits imm offset    */
        c = wmma_f32(a, bv, c);
    }
    /* lane holds 8 consecutive tau for its column -> 32B contiguous store */
    float* dst = frames + (size_t)colp * 64 + 16 * w + 8 * hs;
    *(v8f*)dst = c;
}

/* ---- overlap-add + envelope normalization ------------------------------- */
__global__ void ola_kernel(const float* __restrict__ frames, float* __restrict__ out) {
    const int idx = blockIdx.x * blockDim.x + threadIdx.x;   /* < B*C*T */
    const int t   = idx & (T_LEN - 1);
    const int bo  = idx >> 14;
    const int tp  = t + 32;
    const int n1  = tp >> 5;
    const int off = tp & 31;
    const int n0  = n1 - 1;
    const float* fb = frames + (size_t)bo * (NFRMP * 64);
    const float v = fb[(size_t)n0 * 64 + off + 32] + fb[(size_t)n1 * 64 + off];
    const float cth = cosf(0.09817477042f * (float)off);     /* 2*pi/64 */
    const float env = 0.5f * (1.0f + cth * cth);             /* w(off)^2 + w(off+32)^2 */
    out[idx] = v / env;
}

extern "C" void kernel_launch(void* const* d_in, const int* in_sizes, int n_in,
                              void* d_out, int out_size, void* d_ws, size_t ws_size,
                              hipStream_t stream) {
    const float* x   = (const float*)d_in[0];
    const float* wgt = (const float*)d_in[1];
    float* out = (float*)d_out;
    float* ws  = (float*)d_ws;

    float* A1 = ws + OFF_A1;
    float* A3 = ws + OFF_A3;
    float* Wt = ws + OFF_WT;
    float* U  = ws + OFF_U;
    float* Y  = ws + OFF_Y;
    float* FR = ws + OFF_FR;
    float* XP = ws + OFF_XP;

    prep_kernel<<<64, 256, 0, stream>>>(wgt, A1, A3, Wt);
    pad_kernel<<<dim3((TP_LEN + 255) / 256, NB * NCH), 256, 0, stream>>>(x, XP);
    stft_gemm<<<COL1P / 16, 128, 0, stream>>>(XP, A1, U);
    mix_gemm<<<dim3(COL2P / 16, NFREQ, 2), 128, 0, stream>>>(Wt, U, Y);
    istft_gemm<<<COL1P / 16, 128, 0, stream>>>(A3, Y, FR);
    ola_kernel<<<(NB * NCH * T_LEN) / 256, 256, 0, stream>>>(FR, out);
}